// KOSS_34265249087565
// MI455X (gfx1250) — compile-verified
//
#include <hip/hip_runtime.h>
#include <math.h>

// ---------------------------------------------------------------------------
// Mamba-like SSM forward for MI455X (gfx1250, wave32).
// All heavy GEMMs use V_WMMA_F32_16X16X4_F32 (exact fp32 matrix-core math).
// Each wave computes a 16x64 output tile with 4 independent accumulator
// chains and software double-buffering (fragments for step k+4 are fetched
// before the WMMAs for step k issue, hiding L0/L2 latency behind the matrix
// pipe). N-edge handled by clamped loads; stores predicated only.
// Sequential segment scan (100 steps/layer) is host-orchestrated; kproj
// weights (~30MB) stay resident in the 192MB L2 across the 100 steps.
// ---------------------------------------------------------------------------

#define B_    4
#define L_    2000
#define D_    1024
#define BL_   (B_*L_)
#define SEG_  20
#define NSEG_ (L_/SEG_)
#define DST_  2
#define DCK_  66          // DT_RANK + D_STATE (only used part of xp_w3 output)
#define LD3_  1090        // full xp_w3 leading dim

typedef float v2f __attribute__((ext_vector_type(2)));
typedef float v8f __attribute__((ext_vector_type(8)));

#if __has_builtin(__builtin_amdgcn_wmma_f32_16x16x4_f32)
#define HAVE_WMMA_F32 1
#else
#define HAVE_WMMA_F32 0
#endif

// ---------------------------------------------------------------------------
// Out[M,N](ldo) = act( A[M,K](lda) @ W[K,N](ldw) + bias + (acc ? Out : 0) )
// Wave tile: 16 rows x 64 cols (4 WMMA accumulators). Block: 4 waves = 16x256.
// M must be a multiple of 16 (all call sites: 8000 and 80). K multiple of 4.
// act: 0=none 1=relu 2=softplus
// ---------------------------------------------------------------------------
__global__ void gemm_f32_wmma(const float* __restrict__ A, int lda,
                              const float* __restrict__ W, int ldw,
                              const float* __restrict__ bias,
                              float* __restrict__ Out, int ldo,
                              int K, int N, int act, int acc)
{
  const int lane  = threadIdx.x & 31;
  const int wv    = threadIdx.x >> 5;
  const int lr    = lane & 15;        // row (A/D) or col (B/D) within tile
  const int hf    = lane >> 4;        // lane half selects K sub-pair
  const int m0    = blockIdx.y * 16;
  const int nbase = blockIdx.x * 256 + wv * 64;

  // Clamped columns: an out-of-range B column only pollutes the matching D
  // column, which is never stored -> unconditional loads are safe.
  int cc[4];
  #pragma unroll
  for (int j = 0; j < 4; ++j) {
    int col = nbase + j * 16 + lr;
    cc[j] = (col < N) ? col : (N - 1);
  }

  v8f cacc[4];
  #pragma unroll
  for (int j = 0; j < 4; ++j) cacc[j] = (v8f){0.f,0.f,0.f,0.f,0.f,0.f,0.f,0.f};

  // A-frag (32-bit A 16x4): VGPR0 = K0(half0)/K2(half1), VGPR1 = K1/K3.
  // B-frag (4x16, rows striped over lanes): VGPR v, half h -> row v+2h.
  const float* Apt = A + (size_t)(m0 + lr) * lda + 2 * hf;
  const float* Wpt = W + (size_t)(2 * hf) * ldw;

#if HAVE_WMMA_F32
  auto loadfrag = [&](int k, v2f& a, v2f (&b)[4]) {
    a.x = Apt[k];
    a.y = Apt[k + 1];
    const float* wr0 = Wpt + (size_t)k * ldw;
    const float* wr1 = wr0 + ldw;
    #pragma unroll
    for (int j = 0; j < 4; ++j) {
      b[j].x = wr0[cc[j]];
      b[j].y = wr1[cc[j]];
    }
  };

  v2f a_cur; v2f b_cur[4];
  loadfrag(0, a_cur, b_cur);
  for (int k = 0; k < K; k += 4) {
    // Prefetch next step's fragments (clamped: last iter reloads harmlessly)
    const int kn = (k + 4 < K) ? (k + 4) : k;
    v2f a_nxt; v2f b_nxt[4];
    loadfrag(kn, a_nxt, b_nxt);
    #pragma unroll
    for (int j = 0; j < 4; ++j)
      cacc[j] = __builtin_amdgcn_wmma_f32_16x16x4_f32(false, a_cur, false,
                                                      b_cur[j], (short)0,
                                                      cacc[j], false, false);
    a_cur = a_nxt;
    #pragma unroll
    for (int j = 0; j < 4; ++j) b_cur[j] = b_nxt[j];
  }
#else
  for (int k = 0; k < K; ++k) {
    #pragma unroll
    for (int j = 0; j < 4; ++j) {
      float bv = W[(size_t)k * ldw + cc[j]];
      #pragma unroll
      for (int r = 0; r < 8; ++r)
        cacc[j][r] += A[(size_t)(m0 + r + 8*hf) * lda + k] * bv;
    }
  }
#endif

  #pragma unroll
  for (int j = 0; j < 4; ++j) {
    int col = nbase + j * 16 + lr;
    if (col < N) {
      float bv = bias ? bias[col] : 0.f;
      #pragma unroll
      for (int r = 0; r < 8; ++r) {
        int row = m0 + r + 8 * hf;     // C/D layout: VGPR r -> M=r / M=r+8
        size_t oi = (size_t)row * ldo + col;
        float v = cacc[j][r] + bv;
        if (acc) v += Out[oi];
        if (act == 1)      v = v > 0.f ? v : 0.f;
        else if (act == 2) v = (v > 20.f) ? v : log1pf(expf(v));
        Out[oi] = v;
      }
    }
  }
}

// ---------------------------------------------------------------------------
__global__ void rmsnorm_k(const float* __restrict__ x, const float* __restrict__ w,
                          float* __restrict__ out)
{
  __shared__ float red[256];
  const int r = blockIdx.x;
  const float* xr = x + (size_t)r * D_;
  float s = 0.f;
  for (int i = threadIdx.x; i < D_; i += 256) { float v = xr[i]; s += v*v; }
  red[threadIdx.x] = s; __syncthreads();
  for (int off = 128; off > 0; off >>= 1) {
    if (threadIdx.x < off) red[threadIdx.x] += red[threadIdx.x + off];
    __syncthreads();
  }
  const float rms = rsqrtf(red[0] * (1.f/D_) + 1e-5f);
  for (int i = threadIdx.x; i < D_; i += 256)
    out[(size_t)r * D_ + i] = xr[i] * rms * w[i];
}

// causal depthwise conv, K=4 (no activation)
__global__ void dwconv_k(const float* __restrict__ X, int ldx,
                         const float* __restrict__ w, const float* __restrict__ bias,
                         float* __restrict__ out)
{
  int idx = blockIdx.x * blockDim.x + threadIdx.x;
  if (idx >= BL_ * D_) return;
  int d  = idx & (D_ - 1);
  int bl = idx >> 10;
  int l  = bl % L_;
  int b  = bl / L_;
  float acc = bias[d];
  #pragma unroll
  for (int k = 0; k < 4; ++k) {
    int ls = l - 3 + k;
    if (ls >= 0) acc += X[(size_t)(b * L_ + ls) * ldx + d] * w[d * 4 + k];
  }
  out[(size_t)bl * D_ + d] = acc;
}

// causal depthwise conv, K=4, fused SiLU
__global__ void dwconv_silu_k(const float* __restrict__ X, int ldx,
                              const float* __restrict__ w, const float* __restrict__ bias,
                              float* __restrict__ out)
{
  int idx = blockIdx.x * blockDim.x + threadIdx.x;
  if (idx >= BL_ * D_) return;
  int d  = idx & (D_ - 1);
  int bl = idx >> 10;
  int l  = bl % L_;
  int b  = bl / L_;
  float acc = bias[d];
  #pragma unroll
  for (int k = 0; k < 4; ++k) {
    int ls = l - 3 + k;
    if (ls >= 0) acc += X[(size_t)(b * L_ + ls) * ldx + d] * w[d * 4 + k];
  }
  acc = acc / (1.f + expf(-acc));      // silu(x) = x * sigmoid(x)
  out[(size_t)bl * D_ + d] = acc;
}

// dA[bl,d,n] = exp(delta[bl,d] * -exp(A_log[d,n]))
__global__ void deltaA_k(const float* __restrict__ delta,
                         const float* __restrict__ A_log,
                         float* __restrict__ dA)
{
  int idx = blockIdx.x * blockDim.x + threadIdx.x;
  if (idx >= BL_ * D_) return;
  int d = idx & (D_ - 1);
  float dl = delta[idx];
  #pragma unroll
  for (int n = 0; n < DST_; ++n) {
    float a = -expf(A_log[d * DST_ + n]);
    dA[(size_t)idx * DST_ + n] = expf(dl * a);
  }
}

// per segment: u = diff/(1+diff), diff = (x - <h_prev,C>)^2
__global__ void seg_pre_k(const float* __restrict__ hprev,
                          const float* __restrict__ dck,
                          const float* __restrict__ xc,
                          int t, float* __restrict__ ubuf)
{
  int idx = blockIdx.x * blockDim.x + threadIdx.x;   // B*SEG*D
  if (idx >= B_ * SEG_ * D_) return;
  int d  = idx & (D_ - 1);
  int bs = idx >> 10;                 // b*SEG + s
  int s  = bs % SEG_;
  int b  = bs / SEG_;
  int l  = t * SEG_ + s;
  const float* Cp = dck + (size_t)(b * L_ + l) * DCK_ + 64;
  float hobs = hprev[(size_t)idx * 2 + 0] * Cp[0]
             + hprev[(size_t)idx * 2 + 1] * Cp[1];
  float xv = xc[(size_t)(b * L_ + l) * D_ + d];
  float diff = (xv - hobs) * (xv - hobs);
  ubuf[idx] = diff / (1.f + diff);
}

// FDU (20-pt DFT with LDS twiddles) + in-segment associative scan.
__global__ void seg_scan_k(const float* __restrict__ xc,
                           const float* __restrict__ delta,
                           const float* __restrict__ dA,
                           const float* __restrict__ dck,
                           const float* __restrict__ errK,
                           float* __restrict__ hprev,
                           float* __restrict__ ybuf, int t)
{
  __shared__ float twr[SEG_], twi[SEG_];
  if (threadIdx.x < SEG_) {
    float ang = -6.2831853071795864f * (float)threadIdx.x / (float)SEG_;
    twr[threadIdx.x] = cosf(ang);
    twi[threadIdx.x] = sinf(ang);
  }
  __syncthreads();

  int idx = blockIdx.x * blockDim.x + threadIdx.x;   // B*D = 4096 threads
  int d = idx & (D_ - 1);
  int b = idx >> 10;
  int l0 = t * SEG_;

  float xs[SEG_], fdu[SEG_];
  #pragma unroll
  for (int s = 0; s < SEG_; ++s) {
    xs[s]  = xc[(size_t)(b * L_ + l0 + s) * D_ + d];
    fdu[s] = 0.f;
  }

  // Gaussian FDU kernel taps (K=4, sigma=0.5), normalized.
  const float e2 = expf(-2.f), e8 = expf(-8.f);
  const float ks = 1.f + 2.f * e2 + e8;
  const float ker0 = e8 / ks, ker1 = e2 / ks, ker2 = 1.f / ks, ker3 = e2 / ks;

  for (int k = 0; k < SEG_; ++k) {
    float ar = 0.f, ai = 0.f;
    #pragma unroll
    for (int s = 0; s < SEG_; ++s) {
      int m = (s * k) % SEG_;
      ar += xs[s] * twr[m];
      ai += xs[s] * twi[m];
    }
    // FFT of zero-padded 4-tap kernel at bin k
    float kr = ker0, ki = 0.f;
    { int m = (1 * k) % SEG_; kr += ker1 * twr[m]; ki += ker1 * twi[m]; }
    { int m = (2 * k) % SEG_; kr += ker2 * twr[m]; ki += ker2 * twi[m]; }
    { int m = (3 * k) % SEG_; kr += ker3 * twr[m]; ki += ker3 * twi[m]; }
    float fr = (k < SEG_/2) ? (float)k / SEG_ : (float)(k - SEG_) / SEG_;
    float dt = delta[(size_t)(b * L_ + l0 + k) * D_ + d];
    float w  = 6.2831853071795864f * fr / (dt + 1e-6f);
    float pr = ar * kr - ai * ki;
    float pi = ar * ki + ai * kr;
    float Zr = -w * pi, Zi = w * pr;              // (i*w) * (X*kf)
    #pragma unroll
    for (int s = 0; s < SEG_; ++s) {              // Re(IFFT): conj twiddle
      int m = (s * k) % SEG_;
      fdu[s] += Zr * twr[m] + Zi * twi[m];
    }
  }

  float h0 = 0.f, h1 = 0.f;
  for (int s = 0; s < SEG_; ++s) {
    size_t gi  = (size_t)(b * L_ + l0 + s) * D_ + d;
    size_t bsd = (size_t)((b * SEG_ + s) * D_ + d);
    float er  = errK[bsd];
    float Kdy = er * (fdu[s] * (1.f / SEG_));
    const float* Cp = dck + (size_t)(b * L_ + l0 + s) * DCK_ + 64;
    float C0 = Cp[0], C1 = Cp[1];
    float xv = xs[s];
    // n = 0
    float a  = dA[gi * 2 + 0];
    float KC = er * C0, KCA = KC * a, AK = a - KCA;
    float Ap = AK + AK * KC;
    float Bx = (KCA - a) * xv + Kdy;
    h0 = Ap * h0 + Bx;
    // n = 1
    a  = dA[gi * 2 + 1];
    KC = er * C1; KCA = KC * a; AK = a - KCA;
    Ap = AK + AK * KC;
    Bx = (KCA - a) * xv + Kdy;
    h1 = Ap * h1 + Bx;

    hprev[bsd * 2 + 0] = h0;
    hprev[bsd * 2 + 1] = h1;
    ybuf[gi] = h0 * C0 + h1 * C1;
  }
}

// g = (y + D*xc) * silu(z)
__global__ void gate_k(const float* __restrict__ ybuf, const float* __restrict__ xc,
                       const float* __restrict__ Dp, const float* __restrict__ xz,
                       float* __restrict__ g)
{
  int idx = blockIdx.x * blockDim.x + threadIdx.x;
  if (idx >= BL_ * D_) return;
  int d  = idx & (D_ - 1);
  int bl = idx >> 10;
  float z = xz[(size_t)bl * (2 * D_) + D_ + d];
  float y = ybuf[idx] + Dp[d] * xc[idx];
  g[idx] = y * (z / (1.f + expf(-z)));
}

// ---------------------------------------------------------------------------
static void launch_gemm(hipStream_t st, const float* A, int lda,
                        const float* W, int ldw, const float* bias,
                        float* Out, int ldo, int M, int K, int N,
                        int act, int acc)
{
  dim3 g((unsigned)((N + 255) / 256), (unsigned)(M / 16));
  gemm_f32_wmma<<<g, dim3(128), 0, st>>>(A, lda, W, ldw, bias, Out, ldo,
                                         K, N, act, acc);
}

extern "C" void kernel_launch(void* const* d_in, const int* in_sizes, int n_in,
                              void* d_out, int out_size, void* d_ws, size_t ws_size,
                              hipStream_t stream)
{
  (void)in_sizes; (void)n_in; (void)out_size; (void)ws_size;

  const float* x_in   = (const float*)d_in[0];
  const float* norm_w = (const float*)d_in[1];
  const float* in_w   = (const float*)d_in[2];
  const float* c1_w   = (const float*)d_in[3];
  const float* c1_b   = (const float*)d_in[4];
  const float* c2_w   = (const float*)d_in[5];
  const float* c2_b   = (const float*)d_in[6];
  const float* xp_w1  = (const float*)d_in[7];
  const float* xp_b1  = (const float*)d_in[8];
  const float* xp_w2  = (const float*)d_in[9];
  const float* xp_b2  = (const float*)d_in[10];
  const float* xp_w3  = (const float*)d_in[11];
  const float* xp_b3  = (const float*)d_in[12];
  const float* dt_w   = (const float*)d_in[13];
  const float* dt_b   = (const float*)d_in[14];
  const float* A_log  = (const float*)d_in[15];
  const float* D_par  = (const float*)d_in[16];
  const float* kp_w1  = (const float*)d_in[17];
  const float* kp_b1  = (const float*)d_in[18];
  const float* kp_w2  = (const float*)d_in[19];
  const float* kp_b2  = (const float*)d_in[20];
  const float* kp_w3  = (const float*)d_in[21];
  const float* kp_b3  = (const float*)d_in[22];
  const float* out_w  = (const float*)d_in[23];

  // ---- workspace layout (floats) ----
  float* ws    = (float*)d_ws;
  const size_t BLD = (size_t)BL_ * D_;
  float* xres  = ws;                  // running residual
  float* xn    = xres  + BLD;
  float* xz    = xn    + BLD;         // BL x 2048
  float* xc1   = xz    + 2 * BLD;
  float* xc    = xc1   + BLD;         // silu(conv2) = SSM input
  float* h1    = xc    + BLD;         // BL x 3072
  float* h2    = h1    + 3 * BLD;
  float* dck   = h2    + BLD;         // BL x 66  (delta_raw | C)
  float* deltaB= dck   + (size_t)BL_ * DCK_;   // BL x 1024
  float* dAbuf = deltaB+ BLD;         // BL x 1024 x 2
  float* ybuf  = dAbuf + 2 * BLD;
  float* errK  = ybuf  + BLD;         // B*SEG*D
  float* hprev = errK  + (size_t)B_ * SEG_ * D_;        // *2 states
  float* ubuf  = hprev + (size_t)B_ * SEG_ * D_ * 2;
  float* kh1   = ubuf  + (size_t)B_ * SEG_ * D_;        // 80 x 3072
  float* kh2   = kh1   + (size_t)B_ * SEG_ * 3 * D_;

  const int EW_BLOCKS = (int)((BLD + 255) / 256);

  hipMemcpyAsync(xres, x_in, BLD * sizeof(float),
                 hipMemcpyDeviceToDevice, stream);

  for (int i = 0; i < 4; ++i) {
    const float* nw  = norm_w + (size_t)i * D_;
    const float* iw  = in_w   + (size_t)i * D_ * 2 * D_;
    const float* w1  = xp_w1  + (size_t)i * D_ * 3 * D_;
    const float* b1  = xp_b1  + (size_t)i * 3 * D_;
    const float* w2  = xp_w2  + (size_t)i * 3 * D_ * D_;
    const float* b2  = xp_b2  + (size_t)i * D_;
    const float* w3  = xp_w3  + (size_t)i * D_ * LD3_;
    const float* b3  = xp_b3  + (size_t)i * LD3_;
    const float* dw  = dt_w   + (size_t)i * 64 * D_;
    const float* db  = dt_b   + (size_t)i * D_;
    const float* al  = A_log  + (size_t)i * D_ * DST_;
    const float* dp  = D_par  + (size_t)i * D_;
    const float* kw1 = kp_w1  + (size_t)i * D_ * 3 * D_;
    const float* kb1 = kp_b1  + (size_t)i * 3 * D_;
    const float* kw2 = kp_w2  + (size_t)i * 3 * D_ * D_;
    const float* kb2 = kp_b2  + (size_t)i * D_;
    const float* kw3 = kp_w3  + (size_t)i * D_ * D_;
    const float* kb3 = kp_b3  + (size_t)i * D_;
    const float* ow  = out_w  + (size_t)i * D_ * D_;

    // 1) rmsnorm
    rmsnorm_k<<<BL_, 256, 0, stream>>>(xres, nw, xn);
    // 2) in_proj: xn(8000x1024) @ iw(1024x2048) -> xz
    launch_gemm(stream, xn, D_, iw, 2 * D_, nullptr, xz, 2 * D_,
                BL_, D_, 2 * D_, 0, 0);
    // 3) conv1 (xc half of xz) -> xc1
    dwconv_k<<<EW_BLOCKS, 256, 0, stream>>>(xz, 2 * D_,
        c1_w + (size_t)i * D_ * 4, c1_b + (size_t)i * D_, xc1);
    // 4) conv2 + silu -> xc
    dwconv_silu_k<<<EW_BLOCKS, 256, 0, stream>>>(xc1, D_,
        c2_w + (size_t)i * D_ * 4, c2_b + (size_t)i * D_, xc);
    // 5-7) xp MLP (relu,relu) then dCK (only first 66 cols needed)
    launch_gemm(stream, xc, D_, w1, 3 * D_, b1, h1, 3 * D_, BL_, D_, 3 * D_, 1, 0);
    launch_gemm(stream, h1, 3 * D_, w2, D_, b2, h2, D_, BL_, 3 * D_, D_, 1, 0);
    launch_gemm(stream, h2, D_, w3, LD3_, b3, dck, DCK_, BL_, D_, DCK_, 0, 0);
    // 8) delta = softplus(dck[:, :64] @ dt_w + dt_b)
    launch_gemm(stream, dck, DCK_, dw, D_, db, deltaB, D_, BL_, 64, D_, 2, 0);
    // 9) deltaA
    deltaA_k<<<EW_BLOCKS, 256, 0, stream>>>(deltaB, al, dAbuf);
    // 10) zero carry state
    hipMemsetAsync(errK,  0, (size_t)B_ * SEG_ * D_ * sizeof(float), stream);
    hipMemsetAsync(hprev, 0, (size_t)B_ * SEG_ * D_ * 2 * sizeof(float), stream);

    // 11) sequential segment scan
    for (int t = 0; t < NSEG_; ++t) {
      seg_pre_k<<<(B_ * SEG_ * D_) / 256, 256, 0, stream>>>(hprev, dck, xc, t, ubuf);
      launch_gemm(stream, ubuf, D_, kw1, 3 * D_, kb1, kh1, 3 * D_,
                  B_ * SEG_, D_, 3 * D_, 1, 0);
      launch_gemm(stream, kh1, 3 * D_, kw2, D_, kb2, kh2, D_,
                  B_ * SEG_, 3 * D_, D_, 1, 0);
      launch_gemm(stream, kh2, D_, kw3, D_, kb3, errK, D_,
                  B_ * SEG_, D_, D_, 0, 1);                 // errK += kproj
      seg_scan_k<<<(B_ * D_) / 256, 256, 0, stream>>>(xc, deltaB, dAbuf, dck,
                                                      errK, hprev, ybuf, t);
    }

    // 12) gate: g = (y + D*xc) * silu(z)   (reuse h2)
    gate_k<<<EW_BLOCKS, 256, 0, stream>>>(ybuf, xc, dp, xz, h2);
    // 13) residual: xres += g @ out_w
    launch_gemm(stream, h2, D_, ow, D_, nullptr, xres, D_, BL_, D_, D_, 0, 1);
  }

  hipMemcpyAsync(d_out, xres, BLD * sizeof(float),
                 hipMemcpyDeviceToDevice, stream);
}